// Canny_12120397709496
// MI455X (gfx1250) — compile-verified
//
#include <hip/hip_runtime.h>
#include <math.h>

// CDNA5 wave32 WMMA fragment types: 16x16x4 f32 -> A,B = 2 VGPR (v2f), C/D = 8 VGPR (v8f)
typedef __attribute__((ext_vector_type(2))) float v2f;
typedef __attribute__((ext_vector_type(8))) float v8f;

// Normalized 5-tap Gaussian, sigma=1 (matches reference GAUSS 1-D factor).
// Branchless: chain of v_cndmask selects, no EXEC manipulation (keeps the
// WMMA operand setup convergent; WMMA requires EXEC == all ones).
__device__ __forceinline__ float g5w(int t) {
    const unsigned ut = (unsigned)t;
    float w = 0.0f;
    w = (ut == 0u) ? 0.054488685f : w;
    w = (ut == 1u) ? 0.244201342f : w;
    w = (ut == 2u) ? 0.402619947f : w;
    w = (ut == 3u) ? 0.244201342f : w;
    w = (ut == 4u) ? 0.054488685f : w;
    return w;
}

// ---------------------------------------------------------------------------
// Pass 1: gray conversion fused with horizontal 5-tap blur (reflect pad in x).
// One wave computes one 16x16 tile: Out(16x16) = In(16x20) x Band(20x16),
// as five V_WMMA_F32_16X16X4_F32 with K-slices of 4.
// ---------------------------------------------------------------------------
__global__ void __launch_bounds__(256) k_hblur(const float* __restrict__ x,
                                               float* __restrict__ hb,
                                               int N, int H, int W) {
    __shared__ float lds[8][16][21];  // stride 21: gcd(21,64)=1 -> conflict-free
    const int tid  = threadIdx.x;
    const int wv   = tid >> 5;
    const int lane = tid & 31;
    const int t    = blockIdx.x * 8 + wv;       // tile id
    const int n    = t >> 12;                   // 4096 tiles per image (64x64)
    const int rem  = t & 4095;
    const int y0   = (rem >> 6) << 4;
    const int x0   = (rem & 63) << 4;
    const size_t HW = (size_t)H * W;
    const float* r0 = x + (size_t)n * 3 * HW;
    const float* r1 = r0 + HW;
    const float* r2 = r1 + HW;

    // Fill 16x20 gray tile (reflect pad in x), 10 elements per lane.
    #pragma unroll
    for (int e = 0; e < 10; ++e) {
        int i  = lane + 32 * e;       // 0..319
        int rr = i / 20, cc = i % 20;
        int gx = x0 - 2 + cc;
        gx = gx < 0 ? -gx : (gx >= W ? 2 * W - 2 - gx : gx);
        size_t off = (size_t)(y0 + rr) * W + gx;
        lds[wv][rr][cc] = 0.299f * r0[off] + 0.587f * r1[off] + 0.114f * r2[off];
    }
    __syncthreads();

    const int m  = lane & 15;   // A row / B,C,D column
    const int hf = lane >> 4;   // lane-half selects K (A/B) or M+8 (C/D)
    v8f acc = {};
    #pragma unroll
    for (int kb = 0; kb < 5; ++kb) {
        const int K0 = kb * 4 + 2 * hf;
        v2f a, b;
        a.x = lds[wv][m][K0];       // A[m, K0]
        a.y = lds[wv][m][K0 + 1];   // A[m, K0+1]
        b.x = g5w(K0 - m);          // Band[K0, j=m] = g[K0-j]
        b.y = g5w(K0 + 1 - m);
        acc = __builtin_amdgcn_wmma_f32_16x16x4_f32(false, a, false, b,
                                                    (short)0, acc, false, false);
    }
    union { v8f v; float f[8]; } u; u.v = acc;
    float* o = hb + (size_t)n * HW;
    #pragma unroll
    for (int i = 0; i < 8; ++i)
        o[(size_t)(y0 + i + 8 * hf) * W + (x0 + m)] = u.f[i];
}

// ---------------------------------------------------------------------------
// Pass 2: vertical 5-tap blur (reflect pad in y): Out = Band(16x20) x In(20x16).
// ---------------------------------------------------------------------------
__global__ void __launch_bounds__(256) k_vblur(const float* __restrict__ hb,
                                               float* __restrict__ bl,
                                               int N, int H, int W) {
    __shared__ float lds[8][20][16];  // row stride 16: 16 consecutive lanes -> no conflict
    const int tid  = threadIdx.x;
    const int wv   = tid >> 5;
    const int lane = tid & 31;
    const int t    = blockIdx.x * 8 + wv;
    const int n    = t >> 12;
    const int rem  = t & 4095;
    const int y0   = (rem >> 6) << 4;
    const int x0   = (rem & 63) << 4;
    const size_t HW = (size_t)H * W;
    const float* src = hb + (size_t)n * HW;

    #pragma unroll
    for (int e = 0; e < 10; ++e) {
        int i  = lane + 32 * e;       // 0..319
        int rr = i / 16, cc = i % 16; // 20 rows x 16 cols
        int yy = y0 - 2 + rr;
        yy = yy < 0 ? -yy : (yy >= H ? 2 * H - 2 - yy : yy);
        lds[wv][rr][cc] = src[(size_t)yy * W + (x0 + cc)];
    }
    __syncthreads();

    const int m  = lane & 15;
    const int hf = lane >> 4;
    v8f acc = {};
    #pragma unroll
    for (int kb = 0; kb < 5; ++kb) {
        const int K0 = kb * 4 + 2 * hf;
        v2f a, b;
        a.x = g5w(K0 - m);          // Band[m, K0] = g[K0-m]
        a.y = g5w(K0 + 1 - m);
        b.x = lds[wv][K0][m];       // In[K0, j=m]
        b.y = lds[wv][K0 + 1][m];
        acc = __builtin_amdgcn_wmma_f32_16x16x4_f32(false, a, false, b,
                                                    (short)0, acc, false, false);
    }
    union { v8f v; float f[8]; } u; u.v = acc;
    float* o = bl + (size_t)n * HW;
    #pragma unroll
    for (int i = 0; i < 8; ++i)
        o[(size_t)(y0 + i + 8 * hf) * W + (x0 + m)] = u.f[i];
}

// ---------------------------------------------------------------------------
// Sobel (edge-clamp pad) + magnitude + quantized gradient direction.
// ---------------------------------------------------------------------------
__global__ void __launch_bounds__(256) k_sobel(const float* __restrict__ blr,
                                               float* __restrict__ mag,
                                               unsigned char* __restrict__ dir,
                                               int N, int H, int W) {
    const int total = N * H * W;
    const int idx = blockIdx.x * 256 + threadIdx.x;
    if (idx >= total) return;
    const int HW = H * W;
    const int n = idx / HW, rem = idx % HW;
    const int y = rem / W, xx = rem % W;
    const float* b = blr + (size_t)n * HW;
    const int ym = y > 0 ? y - 1 : 0, yp = y < H - 1 ? y + 1 : y;
    const int xm = xx > 0 ? xx - 1 : 0, xp = xx < W - 1 ? xx + 1 : xx;
    const float tl = b[(size_t)ym * W + xm], tm = b[(size_t)ym * W + xx], tr = b[(size_t)ym * W + xp];
    const float ml = b[(size_t)y  * W + xm],                              mr = b[(size_t)y  * W + xp];
    const float bL = b[(size_t)yp * W + xm], bm = b[(size_t)yp * W + xx], br = b[(size_t)yp * W + xp];
    const float gx = (tr - tl) + 2.0f * (mr - ml) + (br - bL);
    const float gy = (bL - tl) + 2.0f * (bm - tm) + (br - tr);
    mag[idx] = sqrtf(gx * gx + gy * gy + 1e-6f);
    // round(angle_deg/45) with round-half-even == rint(atan2 * 4/pi)
    const int k = (int)rintf(atan2f(gy, gx) * 1.27323954473516f);  // k in [-4,4]
    dir[idx] = (unsigned char)((k + 8) & 7);
}

// ---------------------------------------------------------------------------
// Directional NMS (zero pad) + double threshold. Writes final magnitude output
// and the initial hysteresis state {0, 0.5, 1.0}.
// ---------------------------------------------------------------------------
__global__ void __launch_bounds__(256) k_nms(const float* __restrict__ mag,
                                             const unsigned char* __restrict__ dir,
                                             float* __restrict__ out_mag,
                                             float* __restrict__ edges,
                                             int N, int H, int W) {
    const int total = N * H * W;
    const int idx = blockIdx.x * 256 + threadIdx.x;
    if (idx >= total) return;
    const int HW = H * W;
    const int n = idx / HW, rem = idx % HW;
    const int y = rem / W, xx = rem % W;
    // _OFFS mapping: i -> (dy,dx)
    const int DY[8] = {0, -1, -1, -1, 0, 1, 1, 1};
    const int DX[8] = {1,  1,  0, -1, -1, -1, 0, 1};
    const int p = dir[idx];
    const int dy = DY[p], dx = DX[p];
    const float* mg = mag + (size_t)n * HW;
    float mp = 0.0f, mn = 0.0f;
    int ny = y + dy, nx = xx + dx;
    if (ny >= 0 && ny < H && nx >= 0 && nx < W) mp = mg[(size_t)ny * W + nx];
    ny = y - dy; nx = xx - dx;
    if (ny >= 0 && ny < H && nx >= 0 && nx < W) mn = mg[(size_t)ny * W + nx];
    const float m = mg[(size_t)y * W + xx];
    const float mo = ((m - mp) > 0.0f && (m - mn) > 0.0f) ? m : 0.0f;
    out_mag[idx] = mo;
    edges[idx] = mo > 0.2f ? 1.0f : (mo > 0.1f ? 0.5f : 0.0f);  // high*0.5 + low*0.5
}

// ---------------------------------------------------------------------------
// Hysteresis pass: 32x32 tile (+1 halo) in LDS; promote weak(0.5)->strong(1.0)
// to local convergence, write back. Monotone, so in-place across blocks is safe.
// ---------------------------------------------------------------------------
__global__ void __launch_bounds__(256) k_hyst(float* __restrict__ e, int H, int W) {
    __shared__ float t[34][34];
    __shared__ int changed;
    const int tx0 = blockIdx.x * 32, ty0 = blockIdx.y * 32;
    float* ep = e + (size_t)blockIdx.z * H * W;
    const int tid = threadIdx.y * 32 + threadIdx.x;
    for (int i = tid; i < 34 * 34; i += 256) {
        int r = i / 34, c = i % 34;
        int yy = ty0 + r - 1, xx = tx0 + c - 1;
        t[r][c] = (yy >= 0 && yy < H && xx >= 0 && xx < W) ? ep[(size_t)yy * W + xx] : 0.0f;
    }
    __syncthreads();
    for (int it = 0; it < 48; ++it) {
        if (tid == 0) changed = 0;
        __syncthreads();
        #pragma unroll
        for (int k = 0; k < 4; ++k) {
            const int r = threadIdx.y + 8 * k + 1;
            const int c = threadIdx.x + 1;
            if (t[r][c] == 0.5f) {
                bool s = t[r-1][c-1] == 1.0f || t[r-1][c] == 1.0f || t[r-1][c+1] == 1.0f ||
                         t[r  ][c-1] == 1.0f ||                      t[r  ][c+1] == 1.0f ||
                         t[r+1][c-1] == 1.0f || t[r+1][c] == 1.0f || t[r+1][c+1] == 1.0f;
                if (s) { t[r][c] = 1.0f; changed = 1; }
            }
        }
        __syncthreads();
        const int ch = changed;
        __syncthreads();
        if (!ch) break;
    }
    #pragma unroll
    for (int k = 0; k < 4; ++k) {
        const int r = threadIdx.y + 8 * k + 1;
        const int c = threadIdx.x + 1;
        ep[(size_t)(ty0 + r - 1) * W + (tx0 + c - 1)] = t[r][c];
    }
}

// Final edges output: converged strong mask (reference returns hm at fixpoint).
__global__ void __launch_bounds__(256) k_final(const float* __restrict__ e,
                                               float* __restrict__ out, int total) {
    const int idx = blockIdx.x * 256 + threadIdx.x;
    if (idx >= total) return;
    out[idx] = (e[idx] == 1.0f) ? 1.0f : 0.0f;
}

extern "C" void kernel_launch(void* const* d_in, const int* in_sizes, int n_in,
                              void* d_out, int out_size, void* d_ws, size_t ws_size,
                              hipStream_t stream) {
    const int N = 4, H = 1024, W = 1024;
    const size_t NHW = (size_t)N * H * W;           // 4 Mi elements
    const float* x = (const float*)d_in[0];
    float* out_mag   = (float*)d_out;
    float* out_edges = out_mag + NHW;
    float* bufA = (float*)d_ws;                     // hblur, then raw magnitude
    float* bufB = bufA + NHW;                       // blurred, then edge state
    unsigned char* dirb = (unsigned char*)(bufB + NHW);

    const int tiles = N * (H / 16) * (W / 16);      // 16384 -> 2048 blocks of 8 waves
    k_hblur<<<tiles / 8, 256, 0, stream>>>(x, bufA, N, H, W);
    k_vblur<<<tiles / 8, 256, 0, stream>>>(bufA, bufB, N, H, W);

    const int total = (int)NHW;
    const int g = (total + 255) / 256;
    k_sobel<<<g, 256, 0, stream>>>(bufB, bufA, dirb, N, H, W);
    k_nms<<<g, 256, 0, stream>>>(bufA, dirb, out_mag, bufB, N, H, W);

    dim3 hgrid(W / 32, H / 32, N), hblk(32, 8);
    for (int i = 0; i < 12; ++i)
        k_hyst<<<hgrid, hblk, 0, stream>>>(bufB, H, W);

    k_final<<<g, 256, 0, stream>>>(bufB, out_edges, total);
}